// LSTM2Model_32229434589515
// MI455X (gfx1250) — compile-verified
//
#include <hip/hip_runtime.h>

typedef float v2f __attribute__((ext_vector_type(2)));
typedef float v8f __attribute__((ext_vector_type(8)));

#define B_   8192
#define T_   60
#define F_   158
#define NW_  4          // waves per block
#define KP4_ 40         // F padded to 160 = 40 k-groups of 4

// LDS layout in v2f fragments: BF[(kk*8+n)*32 + lane] = {WT[klo][col], WT[klo+1][col]}
//   klo = kk*4 + 2*(lane>>4), col = n*16 + (lane&15)
#define FR0_ (KP4_ * 8 * 32)           // 10240 frags, layer-1 input weights
#define FRH_ (8 * 8 * 32)              // 2048 frags, each recurrent / layer-2 matrix
// float offsets
#define OFF_BF0 0
#define OFF_BH0 (2 * FR0_)
#define OFF_BI1 (OFF_BH0 + 2 * FRH_)
#define OFF_BH1 (OFF_BI1 + 2 * FRH_)
#define OFF_SCR (OFF_BH1 + 2 * FRH_)
#define SSTR    34                      // scratch row stride (even -> b64 loads, conflict-free)
#define SCR_PW  (2 * 16 * SSTR)         // 1088 floats per wave
#define SMEM_FLOATS (OFF_SCR + NW_ * SCR_PW)

__device__ __forceinline__ float sig_(float x) {
    return __builtin_amdgcn_rcpf(1.0f + __expf(-x));
}
__device__ __forceinline__ float tanh_(float x) {
    return fmaf(2.0f, __builtin_amdgcn_rcpf(1.0f + __expf(-2.0f * x)), -1.0f);
}
__device__ __forceinline__ v8f wmma_f32(v2f a, v2f b, v8f c) {
    return __builtin_amdgcn_wmma_f32_16x16x4_f32(false, a, false, b, (short)0, c, false, false);
}

// one 16x4 A-fragment times a full 4x128 B block: batch the 8 fragment loads
// so the backend can overlap LDS latency with the WMMA stream (partial dscnt waits)
__device__ __forceinline__ void gemm_step(v2f a, const v2f* __restrict__ bp, v8f acc[8]) {
    v2f b[8];
    #pragma unroll
    for (int n = 0; n < 8; ++n) b[n] = bp[n * 32];
    #pragma unroll
    for (int n = 0; n < 8; ++n) acc[n] = wmma_f32(a, b[n], acc[n]);
}

__global__ __launch_bounds__(32 * NW_, 1)
void lstm2_fused(const float* __restrict__ x,
                 const float* __restrict__ Wih0,
                 const float* __restrict__ Whh0,
                 const float* __restrict__ Wih1,
                 const float* __restrict__ Whh1,
                 const float* __restrict__ W1,
                 const float* __restrict__ b1,
                 const float* __restrict__ W2,
                 float* __restrict__ out)
{
    extern __shared__ float smem[];
    v2f* BF0 = (v2f*)(smem + OFF_BF0);
    v2f* BH0 = (v2f*)(smem + OFF_BH0);
    v2f* BI1 = (v2f*)(smem + OFF_BI1);
    v2f* BH1 = (v2f*)(smem + OFF_BH1);

    const int tid   = threadIdx.x;
    const int wave  = tid >> 5;
    const int lane  = tid & 31;
    const int lhalf = lane >> 4;
    const int lcol  = lane & 15;

    // ---- stage weights as pre-laid-out WMMA B-fragments ----
    for (int idx = tid; idx < FR0_; idx += blockDim.x) {
        int l = idx & 31, n = (idx >> 5) & 7, kk = idx >> 8;
        int klo = kk * 4 + ((l >> 4) << 1);
        int col = n * 16 + (l & 15);
        v2f f;
        f.x = (klo     < F_) ? Wih0[col * F_ + klo    ] : 0.0f;
        f.y = (klo + 1 < F_) ? Wih0[col * F_ + klo + 1] : 0.0f;
        BF0[idx] = f;
    }
    for (int idx = tid; idx < FRH_; idx += blockDim.x) {
        int l = idx & 31, n = (idx >> 5) & 7, kk = idx >> 8;
        int klo = kk * 4 + ((l >> 4) << 1);
        int col = n * 16 + (l & 15);
        v2f f0; f0.x = Whh0[col * 32 + klo]; f0.y = Whh0[col * 32 + klo + 1];
        v2f f1; f1.x = Wih1[col * 32 + klo]; f1.y = Wih1[col * 32 + klo + 1];
        v2f f2; f2.x = Whh1[col * 32 + klo]; f2.y = Whh1[col * 32 + klo + 1];
        BH0[idx] = f0; BI1[idx] = f1; BH1[idx] = f2;
    }
    __syncthreads();

    float* hs1 = smem + OFF_SCR + wave * SCR_PW;   // 16x32 tile, row stride 34
    float* hs2 = hs1 + 16 * SSTR;

    const int bb = (blockIdx.x * NW_ + wave) * 16;
    const float* xrow = x + (size_t)(bb + lcol) * (T_ * F_);

    v8f c1t[2] = {}, c2t[2] = {};
    v2f h1f[8] = {}, h2f[8] = {};

    const v2f* bf0 = BF0 + lane;
    const v2f* bh0 = BH0 + lane;
    const v2f* bi1 = BI1 + lane;
    const v2f* bh1 = BH1 + lane;

    for (int t = 0; t < T_; ++t) {
        const float* xt = xrow + t * F_;

        // ================= layer 1 =================
        {
            v8f acc[8] = {};
            for (int kk = 0; kk < KP4_; ++kk) {         // input proj: X_t * W_ih0^T
                const int col = kk * 4 + lhalf * 2;
                v2f a = {};
                if (col < F_) a = *(const v2f*)(xt + col);   // 8B-aligned (F_*4 % 8 == 0)
                gemm_step(a, bf0 + kk * 256, acc);
            }
            #pragma unroll
            for (int kk = 0; kk < 8; ++kk)              // recurrent: h1 * W_hh0^T
                gemm_step(h1f[kk], bh0 + kk * 256, acc);

            #pragma unroll
            for (int j = 0; j < 2; ++j)
                #pragma unroll
                for (int r = 0; r < 8; ++r) {
                    float iv = sig_(acc[j][r]);
                    float fv = sig_(acc[2 + j][r]);
                    float gv = tanh_(acc[4 + j][r]);
                    float ov = sig_(acc[6 + j][r]);
                    float cn = fv * c1t[j][r] + iv * gv;
                    c1t[j][r] = cn;
                    hs1[(r + 8 * lhalf) * SSTR + j * 16 + lcol] = ov * tanh_(cn);
                }
            #pragma unroll
            for (int kk = 0; kk < 8; ++kk)              // C-layout -> A-fragments
                h1f[kk] = *(const v2f*)(hs1 + lcol * SSTR + kk * 4 + lhalf * 2);
        }

        // ================= layer 2 =================
        {
            v8f acc[8] = {};
            #pragma unroll
            for (int kk = 0; kk < 8; ++kk)              // input proj: h1 * W_ih1^T
                gemm_step(h1f[kk], bi1 + kk * 256, acc);
            #pragma unroll
            for (int kk = 0; kk < 8; ++kk)              // recurrent: h2 * W_hh1^T
                gemm_step(h2f[kk], bh1 + kk * 256, acc);

            #pragma unroll
            for (int j = 0; j < 2; ++j)
                #pragma unroll
                for (int r = 0; r < 8; ++r) {
                    float iv = sig_(acc[j][r]);
                    float fv = sig_(acc[2 + j][r]);
                    float gv = tanh_(acc[4 + j][r]);
                    float ov = sig_(acc[6 + j][r]);
                    float cn = fv * c2t[j][r] + iv * gv;
                    c2t[j][r] = cn;
                    hs2[(r + 8 * lhalf) * SSTR + j * 16 + lcol] = ov * tanh_(cn);
                }
            #pragma unroll
            for (int kk = 0; kk < 8; ++kk)
                h2f[kk] = *(const v2f*)(hs2 + lcol * SSTR + kk * 4 + lhalf * 2);
        }
    }

    // ================= head: y = relu(c_n @ W1^T + b1) @ W2^T =================
    #pragma unroll
    for (int j = 0; j < 2; ++j)
        #pragma unroll
        for (int r = 0; r < 8; ++r) {
            hs1[(r + 8 * lhalf) * SSTR + j * 16 + lcol] = c1t[j][r];
            hs2[(r + 8 * lhalf) * SSTR + j * 16 + lcol] = c2t[j][r];
        }
    const float* cs = hs1 + lhalf * (16 * SSTR);   // lane half selects layer
    float yv = 0.0f;
    for (int u = 0; u < 16; ++u) {
        float s = b1[u];
        #pragma unroll
        for (int k = 0; k < 32; ++k) s = fmaf(cs[lcol * SSTR + k], W1[u * 32 + k], s);
        s = fmaxf(s, 0.0f);
        yv = fmaf(s, W2[u], yv);
    }
    out[lhalf * B_ + bb + lcol] = yv;
}

extern "C" void kernel_launch(void* const* d_in, const int* in_sizes, int n_in,
                              void* d_out, int out_size, void* d_ws, size_t ws_size,
                              hipStream_t stream) {
    (void)in_sizes; (void)n_in; (void)d_ws; (void)ws_size; (void)out_size;
    const float* x    = (const float*)d_in[0];
    const float* Wih0 = (const float*)d_in[1];
    const float* Whh0 = (const float*)d_in[2];
    const float* Wih1 = (const float*)d_in[3];
    const float* Whh1 = (const float*)d_in[4];
    const float* W1   = (const float*)d_in[5];
    const float* b1   = (const float*)d_in[6];
    const float* W2   = (const float*)d_in[7];
    float* out = (float*)d_out;

    const size_t smem_bytes = (size_t)SMEM_FLOATS * sizeof(float);   // ~148 KB (CDNA5: 320KB/WG)
    hipFuncSetAttribute(reinterpret_cast<const void*>(lstm2_fused),
                        hipFuncAttributeMaxDynamicSharedMemorySize, (int)smem_bytes);

    dim3 block(32 * NW_);                 // 4 wave32 waves
    dim3 grid(B_ / (16 * NW_));           // 128 blocks, 512 independent waves
    lstm2_fused<<<grid, block, smem_bytes, stream>>>(x, Wih0, Whh0, Wih1, Whh1, W1, b1, W2, out);
}